// _GATLayer_58815282152005
// MI455X (gfx1250) — compile-verified
//
#include <hip/hip_runtime.h>
#include <hip/hip_bf16.h>

// Problem constants (match reference)
#define Bv 2
#define Nv 10000
#define Ev 160000
#define Dv 256           // D_IN == D_OUT == H*DH
#define Hv 8
#define DHv 32
#define Mrows (Bv * Nv)  // 20000 rows of x / h
#define NEG_SLOPE 0.2f
#define LN_EPS 1e-5f

typedef __attribute__((ext_vector_type(2))) float v2f;
typedef __attribute__((ext_vector_type(8))) float v8f;

// ---- ordered-int encoding for float atomic max ----
__device__ __forceinline__ unsigned f2ord(float f) {
  unsigned u = __float_as_uint(f);
  return (u & 0x80000000u) ? ~u : (u | 0x80000000u);
}
__device__ __forceinline__ float ord2f(unsigned u) {
  u = (u & 0x80000000u) ? (u & 0x7FFFFFFFu) : ~u;
  return __uint_as_float(u);
}
#define ORD_NEG_INF 0x007FFFFFu  // f2ord(-inf)

__device__ __forceinline__ float wave_sum32(float v) {
  v += __shfl_xor(v, 16, 32);
  v += __shfl_xor(v, 8, 32);
  v += __shfl_xor(v, 4, 32);
  v += __shfl_xor(v, 2, 32);
  v += __shfl_xor(v, 1, 32);
  return v;
}

// ---------------------------------------------------------------------------
// Kernel 1: h = x @ W  via V_WMMA_F32_16X16X4_F32.
// grid = (Mrows/16, 2), block = 256 (8 waves). Wave w computes the 16x16 tile
// at (m0 = 16*blockIdx.x, n0 = 16*(blockIdx.y*8 + w)).
// ---------------------------------------------------------------------------
__global__ void gat_gemm_wmma(const float* __restrict__ X,
                              const float* __restrict__ W,
                              float* __restrict__ Hout) {
  const int lane = threadIdx.x & 31;
  const int wave = threadIdx.x >> 5;
  const int half = lane >> 4;   // 0: lanes 0-15, 1: lanes 16-31
  const int j    = lane & 15;
  const int m0 = blockIdx.x * 16;
  const int n0 = (blockIdx.y * 8 + wave) * 16;

  const float* __restrict__ xrow = X + (size_t)(m0 + j) * Dv;
  v8f c = {};
#pragma unroll 4
  for (int k = 0; k < Dv; k += 4) {
    v2f a, b;
    // A 16x4 (MxK): lane row j, VGPR v holds K = v + 2*half
    a.x = xrow[k + 2 * half + 0];
    a.y = xrow[k + 2 * half + 1];
    // B 4x16 (KxN): lane col j, VGPR v holds K = v + 2*half
    b.x = W[(size_t)(k + 2 * half + 0) * Dv + n0 + j];
    b.y = W[(size_t)(k + 2 * half + 1) * Dv + n0 + j];
    c = __builtin_amdgcn_wmma_f32_16x16x4_f32(false, a, false, b,
                                              (short)0, c, false, false);
  }
  // D 16x16: VGPR v holds M = v + 8*half, N = j
#pragma unroll
  for (int v = 0; v < 8; ++v) {
    Hout[(size_t)(m0 + v + 8 * half) * Dv + n0 + j] = c[v];
  }
}

// ---------------------------------------------------------------------------
// Kernel 2: init workspace (m_ord = -inf encoding; denom/accum = 0)
// ---------------------------------------------------------------------------
__global__ void gat_init(unsigned* __restrict__ m_ord,
                         float* __restrict__ zero_base, long n_zero) {
  long t = (long)blockIdx.x * blockDim.x + threadIdx.x;
  if (t < n_zero) zero_base[t] = 0.0f;
  if (t < (long)Nv * Bv * Hv) m_ord[t] = ORD_NEG_INF;
}

// ---------------------------------------------------------------------------
// Kernel 3: per-edge logits e = leaky_relu(h_src.a_l + h_dst.a_r),
// atomicMax into per-(dst,b,h) max. One wave per edge, lane = dh.
// ---------------------------------------------------------------------------
__global__ void gat_edge_logits(const float* __restrict__ Hm,
                                const float* __restrict__ a_l,
                                const float* __restrict__ a_r,
                                const int* __restrict__ eidx,
                                float* __restrict__ e_out,
                                unsigned* __restrict__ m_ord) {
  const int lane = threadIdx.x & 31;
  const int wave = threadIdx.x >> 5;
  const long edge = (long)blockIdx.x * 8 + wave;
  if (edge >= Ev) return;
  const int src = eidx[edge];
  const int dst = eidx[Ev + edge];

  __builtin_prefetch(&Hm[(size_t)src * Dv + lane], 0, 1);
  __builtin_prefetch(&Hm[(size_t)dst * Dv + lane], 0, 1);

#pragma unroll
  for (int b = 0; b < Bv; ++b) {
    const float* hs = Hm + (size_t)(b * Nv + src) * Dv;
    const float* hd = Hm + (size_t)(b * Nv + dst) * Dv;
#pragma unroll
    for (int h = 0; h < Hv; ++h) {
      const int d = h * DHv + lane;
      float v = hs[d] * a_l[d] + hd[d] * a_r[d];
      v = wave_sum32(v);
      v = (v > 0.0f) ? v : NEG_SLOPE * v;
      if (lane == 0) {
        e_out[((size_t)edge * Bv + b) * Hv + h] = v;
        atomicMax(&m_ord[((size_t)dst * Bv + b) * Hv + h], f2ord(v));
      }
    }
  }
}

// ---------------------------------------------------------------------------
// Kernel 4: ex = exp(e - m[dst]); denom[dst] += ex. Thread per (edge,b,h).
// ---------------------------------------------------------------------------
__global__ void gat_edge_exp(const int* __restrict__ eidx,
                             float* __restrict__ e_io,
                             const unsigned* __restrict__ m_ord,
                             float* __restrict__ denom) {
  const long t = (long)blockIdx.x * blockDim.x + threadIdx.x;
  if (t >= (long)Ev * Bv * Hv) return;
  const int h = (int)(t % Hv);
  const int b = (int)((t / Hv) % Bv);
  const long edge = t / (Bv * Hv);
  const int dst = eidx[Ev + edge];
  const size_t seg = ((size_t)dst * Bv + b) * Hv + h;
  const float m = ord2f(m_ord[seg]);
  const float ex = __expf(e_io[t] - m);
  e_io[t] = ex;
  unsafeAtomicAdd(&denom[seg], ex);
}

// ---------------------------------------------------------------------------
// Kernel 5: accum[dst] += (ex/denom[dst]) * h_src. One wave per edge.
// ---------------------------------------------------------------------------
__global__ void gat_edge_accum(const float* __restrict__ Hm,
                               const int* __restrict__ eidx,
                               const float* __restrict__ e_exp,
                               const float* __restrict__ denom,
                               float* __restrict__ accum) {
  const int lane = threadIdx.x & 31;
  const int wave = threadIdx.x >> 5;
  const long edge = (long)blockIdx.x * 8 + wave;
  if (edge >= Ev) return;
  const int src = eidx[edge];
  const int dst = eidx[Ev + edge];

  __builtin_prefetch(&Hm[(size_t)src * Dv + lane], 0, 1);

#pragma unroll
  for (int b = 0; b < Bv; ++b) {
    const float* hs = Hm + (size_t)(b * Nv + src) * Dv;
    float* ac = accum + (size_t)(b * Nv + dst) * Dv;
#pragma unroll
    for (int h = 0; h < Hv; ++h) {
      const float attn = e_exp[((size_t)edge * Bv + b) * Hv + h] /
                         denom[((size_t)dst * Bv + b) * Hv + h];
      const int d = h * DHv + lane;
      unsafeAtomicAdd(&ac[d], attn * hs[d]);
    }
  }
}

// ---------------------------------------------------------------------------
// Kernel 6: out = ELU(LayerNorm(accum + x)). One 256-thread block per row.
// ---------------------------------------------------------------------------
__global__ void gat_final_ln(const float* __restrict__ accum,
                             const float* __restrict__ X,
                             const float* __restrict__ gamma,
                             const float* __restrict__ beta,
                             float* __restrict__ out) {
  const int tid = threadIdx.x;
  const int lane = tid & 31;
  const int wave = tid >> 5;
  const size_t row = blockIdx.x;
  const size_t base = row * Dv;

  float val = accum[base + tid] + X[base + tid];

  __shared__ float red[16];
  float s = wave_sum32(val);
  float s2 = wave_sum32(val * val);
  if (lane == 0) { red[wave] = s; red[8 + wave] = s2; }
  __syncthreads();
  if (tid == 0) {
    float a = 0.0f, bq = 0.0f;
#pragma unroll
    for (int i = 0; i < 8; ++i) { a += red[i]; bq += red[8 + i]; }
    red[0] = a; red[8] = bq;
  }
  __syncthreads();
  const float mean = red[0] * (1.0f / Dv);
  const float var = red[8] * (1.0f / Dv) - mean * mean;
  const float inv = rsqrtf(var + LN_EPS);
  float o = (val - mean) * inv * gamma[tid] + beta[tid];
  o = (o > 0.0f) ? o : (__expf(o) - 1.0f);
  out[base + tid] = o;
}

// ---------------------------------------------------------------------------
extern "C" void kernel_launch(void* const* d_in, const int* in_sizes, int n_in,
                              void* d_out, int out_size, void* d_ws, size_t ws_size,
                              hipStream_t stream) {
  const float* x     = (const float*)d_in[0];
  const float* W     = (const float*)d_in[1];
  const float* a_l   = (const float*)d_in[2];
  const float* a_r   = (const float*)d_in[3];
  const float* gamma = (const float*)d_in[4];
  const float* beta  = (const float*)d_in[5];
  const int*   eidx  = (const int*)d_in[6];
  float* out = (float*)d_out;

  // Workspace layout (floats)
  float*    h     = (float*)d_ws;                        // Mrows*Dv   = 5,120,000
  float*    e     = h + (size_t)Mrows * Dv;              // Ev*Bv*Hv   = 2,560,000
  unsigned* m_ord = (unsigned*)(e + (size_t)Ev * Bv * Hv); // Nv*Bv*Hv = 160,000
  float*    denom = (float*)(m_ord + (size_t)Nv * Bv * Hv); // 160,000
  float*    accum = denom + (size_t)Nv * Bv * Hv;          // 5,120,000
  const long n_zero = (long)Nv * Bv * Hv + (long)Mrows * Dv; // denom + accum

  // 1) GEMM h = x @ W (WMMA fp32)
  dim3 ggrid(Mrows / 16, 2);
  gat_gemm_wmma<<<ggrid, 256, 0, stream>>>(x, W, h);

  // 2) init m_ord / denom / accum
  {
    long nthr = n_zero;
    int blocks = (int)((nthr + 255) / 256);
    gat_init<<<blocks, 256, 0, stream>>>(m_ord, denom, n_zero);
  }

  // 3) edge logits + segment max
  gat_edge_logits<<<Ev / 8, 256, 0, stream>>>(h, a_l, a_r, eidx, e, m_ord);

  // 4) exp + segment sum
  {
    long nthr = (long)Ev * Bv * Hv;
    int blocks = (int)((nthr + 255) / 256);
    gat_edge_exp<<<blocks, 256, 0, stream>>>(eidx, e, m_ord, denom);
  }

  // 5) weighted scatter-add
  gat_edge_accum<<<Ev / 8, 256, 0, stream>>>(h, eidx, e, denom, accum);

  // 6) residual + LayerNorm + ELU
  gat_final_ln<<<Mrows, 256, 0, stream>>>(accum, x, gamma, beta, out);
}